// PriorTransformerModel_90099823935926
// MI455X (gfx1250) — compile-verified
//
#include <hip/hip_runtime.h>
#include <hip/hip_bf16.h>
#include <math.h>

// ---------------------------------------------------------------------------
// PriorTransformerModel on MI455X (gfx1250), full-FP32 path via
// V_WMMA_F32_16X16X4_F32 (exact FP32 matrix math, wave32).
//
// Shapes: S=1024 B=4 F=100 D=512 H=4 DH=128 NHID=1024 L=4 P=128 NOUT=10
//         CP=768 T=1152
// ---------------------------------------------------------------------------

typedef float v2f __attribute__((ext_vector_type(2)));
typedef float v8f __attribute__((ext_vector_type(8)));

#define WMMA_F32(a, b, c) \
    __builtin_amdgcn_wmma_f32_16x16x4_f32(false, (a), false, (b), (short)0, (c), false, false)

static constexpr int kS    = 1024;
static constexpr int kB    = 4;
static constexpr int kF    = 100;
static constexpr int kD    = 512;
static constexpr int kH    = 4;
static constexpr int kDH   = 128;
static constexpr int kNHID = 1024;
static constexpr int kL    = 4;
static constexpr int kP    = 128;
static constexpr int kNOUT = 10;
static constexpr int kCP   = 768;
static constexpr int kT    = kS + kP;          // 1152
static constexpr int kTB   = kT * kB;          // 4608
static constexpr int kQKVs = kB * 3 * kD;      // 6144: stride between t rows in qkv buf

// ---------------------------------------------------------------------------
// Generic GEMM:  C[M,N] = act( A[M,K] * W[N,K]^T + bias[N] )
// A row-major (lda), W row-major (ldw) -- matches einsum('mk,nk->mn').
//
// One wave computes a 32x32 register-blocked tile: 2 A fragments x 2 B
// fragments -> 4 independent v8f accumulators. Per K-step (K+=4):
//   4 x global_load_b64 feeds, 4 independent v_wmma_f32_16x16x4_f32
// => 2x less L2 traffic than 16x16 tiling and 4 independent WMMA dep chains
//    (hides the WMMA result latency instead of RAW-stalling one accumulator).
//
// ISA operand layout (CDNA5 ISA, 32-bit A 16x4 / B 4x16 / C 16x16):
//   lane l: half=l>>4, lr=l&15
//   A vgpr pair = A[m=lr][k+2*half], A[m=lr][k+2*half+1]
//   B vgpr pair = W[n=lr][k+2*half], W[n=lr][k+2*half+1]
//   C vgpr r    = (M = r + 8*half, N = lr)
// ---------------------------------------------------------------------------
__global__ __launch_bounds__(32)
void gemm_wmma_f32_kernel(const float* __restrict__ A, const float* __restrict__ W,
                          const float* __restrict__ bias, float* __restrict__ C,
                          int M, int N, int K, int lda, int ldw, int ldc, int act)
{
    const int lane = threadIdx.x;
    const int hl   = lane >> 4;
    const int lr   = lane & 15;
    const int nt   = blockIdx.x;       // 32-wide N block
    const int mt   = blockIdx.y;       // 32-tall M block

    const int m0 = mt * 32 + lr;
    const int m1 = m0 + 16;
    const int n0 = nt * 32 + lr;
    const int n1 = n0 + 16;
    const int am0 = (m0 < M) ? m0 : (M - 1);   // clamped: garbage never stored
    const int am1 = (m1 < M) ? m1 : (M - 1);
    const int wn0 = (n0 < N) ? n0 : (N - 1);
    const int wn1 = (n1 < N) ? n1 : (N - 1);

    const float* __restrict__ arow0 = A + (size_t)am0 * lda + 2 * hl;
    const float* __restrict__ arow1 = A + (size_t)am1 * lda + 2 * hl;
    const float* __restrict__ wrow0 = W + (size_t)wn0 * ldw + 2 * hl;
    const float* __restrict__ wrow1 = W + (size_t)wn1 * ldw + 2 * hl;

    v8f acc00 = {}, acc01 = {}, acc10 = {}, acc11 = {};
    for (int k = 0; k < K; k += 4) {
        const v2f a0 = *(const v2f*)(arow0 + k);
        const v2f a1 = *(const v2f*)(arow1 + k);
        const v2f b0 = *(const v2f*)(wrow0 + k);
        const v2f b1 = *(const v2f*)(wrow1 + k);
        acc00 = WMMA_F32(a0, b0, acc00);
        acc01 = WMMA_F32(a0, b1, acc01);
        acc10 = WMMA_F32(a1, b0, acc10);
        acc11 = WMMA_F32(a1, b1, acc11);
    }

    const float bi0 = (bias != nullptr && n0 < N) ? bias[n0] : 0.0f;
    const float bi1 = (bias != nullptr && n1 < N) ? bias[n1] : 0.0f;

#pragma unroll
    for (int i = 0; i < 2; ++i) {          // M sub-tile
#pragma unroll
        for (int j = 0; j < 2; ++j) {      // N sub-tile
            const v8f acc = (i == 0) ? (j == 0 ? acc00 : acc01)
                                     : (j == 0 ? acc10 : acc11);
            const float bi = (j == 0) ? bi0 : bi1;
            const int   on = nt * 32 + j * 16 + lr;
#pragma unroll
            for (int r = 0; r < 8; ++r) {
                const int om = mt * 32 + i * 16 + r + 8 * hl;
                if (om < M && on < N) {
                    float v = acc[r] + bi;
                    if (act == 1)      v = 0.5f * v * (1.0f + erff(v * 0.70710678118654752f));
                    else if (act == 2) v = fmaxf(v, 0.0f);
                    C[(size_t)om * ldc + on] = v;
                }
            }
        }
    }
}

// ---------------------------------------------------------------------------
// Assemble src[T,B,D]:
//   t <  CP : src += y_src[t,b]*yenc_w[d] + yenc_b[d]   (src already holds enc GEMM)
//   CP<=t<S : leave enc GEMM result
//   t >= S  : src = x_test[t-S, d]  (broadcast over b)
// ---------------------------------------------------------------------------
__global__ void assemble_src_kernel(float* __restrict__ src,
                                    const float* __restrict__ y_src,
                                    const float* __restrict__ yenc_w,
                                    const float* __restrict__ yenc_b,
                                    const float* __restrict__ x_test)
{
    const int idx = blockIdx.x * blockDim.x + threadIdx.x;
    if (idx >= kT * kB * kD) return;
    const int d  = idx & (kD - 1);
    const int tb = idx >> 9;           // kD == 512
    const int t  = tb >> 2;            // kB == 4
    const int b  = tb & 3;
    if (t < kCP) {
        src[idx] += y_src[t * kB + b] * yenc_w[d] + yenc_b[d];
    } else if (t >= kS) {
        src[idx] = x_test[(t - kS) * kD + d];
    }
}

// ---------------------------------------------------------------------------
// Prefix attention, one wave per (query 16-tile, b, h).
// Mask: key j allowed iff j < CP or j == t.  CP=768 is 16-aligned, so:
//   48 full 16x16 prefix score tiles via WMMA, plus (for qt>=48) the query
//   tile's own diagonal tile with off-diagonal entries forced to -1e30.
// Scores staged in LDS (16 x 800 f32 = 50 KB), exact softmax, then ctx = P*V
// again via WMMA (A operand streamed from LDS, B operand = V columns).
// ---------------------------------------------------------------------------
__global__ __launch_bounds__(32)
void attention_kernel(const float* __restrict__ qkv, float* __restrict__ ctxout)
{
    __shared__ float sc[16][800];      // 784 used columns, padded stride

    const int lane = threadIdx.x;
    const int hl   = lane >> 4;
    const int lr   = lane & 15;
    const int qt   = blockIdx.x;       // 0..71  query 16-tile
    const int bh   = blockIdx.y;       // 0..15
    const int b    = bh >> 2;
    const int h    = bh & 3;
    const float scale = 0.08838834764831845f;   // 1/sqrt(DH)

    const float* __restrict__ qbase = qkv + b * (3 * kD) + h * kDH;
    const float* __restrict__ kbase = qbase + kD;
    const float* __restrict__ vbase = qbase + 2 * kD;

    const int  m    = qt * 16 + lr;    // query row (t index) fed by this lane
    const bool diag = (qt >= kCP / 16);
    const int  nkt  = diag ? 49 : 48;

    // ---- scores = (Q K^T) * scale  (+ mask), staged to LDS ----
    for (int kt = 0; kt < nkt; ++kt) {
        const int j0 = (kt < 48) ? kt * 16 : qt * 16;
        const float* __restrict__ arow = qbase + (size_t)m * kQKVs + 2 * hl;
        const float* __restrict__ wrow = kbase + (size_t)(j0 + lr) * kQKVs + 2 * hl;
        v8f acc = {};
        for (int k = 0; k < kDH; k += 4) {
            v2f a  = *(const v2f*)(arow + k);
            v2f bb = *(const v2f*)(wrow + k);
            acc = WMMA_F32(a, bb, acc);
        }
#pragma unroll
        for (int r = 0; r < 8; ++r) {
            const int lm = r + 8 * hl;             // local query row 0..15
            float v = acc[r] * scale;
            if (kt == 48) {                        // diagonal tile: self only
                const int tg = qt * 16 + lm;
                const int j  = j0 + lr;
                if (j != tg) v = -1e30f;
            }
            sc[lm][kt * 16 + lr] = v;
        }
    }
    __syncthreads();

    // ---- exact softmax per query row (lanes 0..15, one row each) ----
    const int nk = diag ? (kCP + 16) : kCP;
    if (lane < 16) {
        float mx = -1e30f;
        for (int j = 0; j < nk; ++j) mx = fmaxf(mx, sc[lane][j]);
        float sum = 0.0f;
        for (int j = 0; j < nk; ++j) {
            const float e = __expf(sc[lane][j] - mx);
            sc[lane][j] = e;
            sum += e;
        }
        const float inv = 1.0f / sum;
        for (int j = 0; j < nk; ++j) sc[lane][j] *= inv;
    }
    __syncthreads();

    // ---- ctx = P * V, 8 output d-tiles of 16 ----
    for (int dt = 0; dt < 8; ++dt) {
        const float* __restrict__ vcol = vbase + dt * 16 + lr;  // output column d
        v8f acc = {};
        for (int k = 0; k < nk; k += 4) {
            const int j = k + 2 * hl;                           // LDS score column
            v2f a = *(const v2f*)&sc[lr][j];
            // map LDS column -> global key index (768..783 are the diagonal tile)
            const int jg = (j < kCP) ? j : qt * 16 + (j - kCP);
            v2f bb;
            bb.x = vcol[(size_t)jg * kQKVs];
            bb.y = vcol[(size_t)(jg + 1) * kQKVs];
            acc = WMMA_F32(a, bb, acc);
        }
#pragma unroll
        for (int r = 0; r < 8; ++r) {
            const int tg = qt * 16 + r + 8 * hl;
            ctxout[(size_t)(tg * kB + b) * kD + h * kDH + dt * 16 + lr] = acc[r];
        }
    }
}

// ---------------------------------------------------------------------------
// src = LayerNorm(src + delta) * g + bta     one wave32 per (t,b) row, D=512
// ---------------------------------------------------------------------------
__global__ __launch_bounds__(32)
void add_layernorm_kernel(float* __restrict__ src, const float* __restrict__ delta,
                          const float* __restrict__ g, const float* __restrict__ bta)
{
    const int row  = blockIdx.x;
    const int lane = threadIdx.x;
    float vals[16];
    float s = 0.0f;
#pragma unroll
    for (int i = 0; i < 16; ++i) {
        const int d = lane + i * 32;
        const float v = src[(size_t)row * kD + d] + delta[(size_t)row * kD + d];
        vals[i] = v;
        s += v;
    }
#pragma unroll
    for (int off = 16; off > 0; off >>= 1) s += __shfl_xor(s, off, 32);
    const float mean = s * (1.0f / (float)kD);

    float vs = 0.0f;
#pragma unroll
    for (int i = 0; i < 16; ++i) { const float dv = vals[i] - mean; vs += dv * dv; }
#pragma unroll
    for (int off = 16; off > 0; off >>= 1) vs += __shfl_xor(vs, off, 32);
    const float rstd = rsqrtf(vs * (1.0f / (float)kD) + 1e-5f);

#pragma unroll
    for (int i = 0; i < 16; ++i) {
        const int d = lane + i * 32;
        src[(size_t)row * kD + d] = (vals[i] - mean) * rstd * g[d] + bta[d];
    }
}

// ---------------------------------------------------------------------------
// Host-side orchestration
// ---------------------------------------------------------------------------
static inline void launch_gemm(const float* A, const float* W, const float* bias,
                               float* C, int M, int N, int K,
                               int lda, int ldw, int ldc, int act, hipStream_t s)
{
    dim3 grid((N + 31) / 32, (M + 31) / 32);
    gemm_wmma_f32_kernel<<<grid, 32, 0, s>>>(A, W, bias, C, M, N, K, lda, ldw, ldc, act);
}

extern "C" void kernel_launch(void* const* d_in, const int* in_sizes, int n_in,
                              void* d_out, int out_size, void* d_ws, size_t ws_size,
                              hipStream_t stream)
{
    (void)in_sizes; (void)n_in; (void)out_size; (void)ws_size;

    const float* x_src  = (const float*)d_in[0];
    const float* y_src  = (const float*)d_in[1];
    const float* enc_w  = (const float*)d_in[2];
    const float* enc_b  = (const float*)d_in[3];
    const float* yenc_w = (const float*)d_in[4];
    const float* yenc_b = (const float*)d_in[5];
    const float* x_test = (const float*)d_in[6];
    const float* qkv_w  = (const float*)d_in[7];
    const float* qkv_b  = (const float*)d_in[8];
    const float* out_w  = (const float*)d_in[9];
    const float* out_b  = (const float*)d_in[10];
    const float* ln1_g  = (const float*)d_in[11];
    const float* ln1_b  = (const float*)d_in[12];
    const float* lin1_w = (const float*)d_in[13];
    const float* lin1_b = (const float*)d_in[14];
    const float* lin2_w = (const float*)d_in[15];
    const float* lin2_b = (const float*)d_in[16];
    const float* ln2_g  = (const float*)d_in[17];
    const float* ln2_b  = (const float*)d_in[18];
    const float* dec_w1 = (const float*)d_in[19];
    const float* dec_b1 = (const float*)d_in[20];
    const float* dec_w2 = (const float*)d_in[21];
    const float* dec_b2 = (const float*)d_in[22];
    const float* dp_w1  = (const float*)d_in[23];
    const float* dp_b1  = (const float*)d_in[24];
    const float* dp_w2  = (const float*)d_in[25];
    const float* dp_b2  = (const float*)d_in[26];

    float* out = (float*)d_out;

    // workspace layout (floats); ff aliases qkv region (dead by FF time)
    float* ws  = (float*)d_ws;
    float* src = ws;                                   // 4608 * 512
    float* qkv = src + (size_t)kTB * kD;               // 4608 * 1536
    float* ctx = qkv + (size_t)kTB * 3 * kD;           // 4608 * 512
    float* tmp = ctx + (size_t)kTB * kD;               // 4608 * 512
    float* ff  = qkv;                                  // 4608 * 1024 (alias)

    // 1) encoder: x = x_src @ enc_w^T + enc_b  -> src rows 0..S*B-1
    launch_gemm(x_src, enc_w, enc_b, src, kS * kB, kD, kF, kF, kF, kD, 0, stream);

    // 2) src assembly (+y for t<CP, x_test rows for t>=S)
    {
        const int n = kT * kB * kD;
        assemble_src_kernel<<<(n + 255) / 256, 256, 0, stream>>>(src, y_src, yenc_w, yenc_b, x_test);
    }

    // 3) transformer layers
    for (int l = 0; l < kL; ++l) {
        const float* qw  = qkv_w  + (size_t)l * 3 * kD * kD;
        const float* qb  = qkv_b  + (size_t)l * 3 * kD;
        const float* ow  = out_w  + (size_t)l * kD * kD;
        const float* ob  = out_b  + (size_t)l * kD;
        const float* g1  = ln1_g  + (size_t)l * kD;
        const float* b1  = ln1_b  + (size_t)l * kD;
        const float* w1  = lin1_w + (size_t)l * kNHID * kD;
        const float* wb1 = lin1_b + (size_t)l * kNHID;
        const float* w2  = lin2_w + (size_t)l * kD * kNHID;
        const float* wb2 = lin2_b + (size_t)l * kD;
        const float* g2  = ln2_g  + (size_t)l * kD;
        const float* b2  = ln2_b  + (size_t)l * kD;

        // qkv = src @ qkv_w^T + qkv_b
        launch_gemm(src, qw, qb, qkv, kTB, 3 * kD, kD, kD, kD, 3 * kD, 0, stream);
        // attention -> ctx
        attention_kernel<<<dim3(kT / 16, kB * kH), 32, 0, stream>>>(qkv, ctx);
        // attn_out = ctx @ out_w^T + out_b
        launch_gemm(ctx, ow, ob, tmp, kTB, kD, kD, kD, kD, kD, 0, stream);
        // src = LN(src + attn_out)
        add_layernorm_kernel<<<kTB, 32, 0, stream>>>(src, tmp, g1, b1);
        // ff = gelu(src @ lin1_w^T + lin1_b)
        launch_gemm(src, w1, wb1, ff, kTB, kNHID, kD, kD, kD, kNHID, 1, stream);
        // tmp = ff @ lin2_w^T + lin2_b
        launch_gemm(ff, w2, wb2, tmp, kTB, kD, kNHID, kNHID, kNHID, kD, 0, stream);
        // src = LN(src + ff_out)
        add_layernorm_kernel<<<kTB, 32, 0, stream>>>(src, tmp, g2, b2);
    }

    // 4) decoders — only rows that survive out[cp:] are computed.
    // dec head: src rows [CP, S) -> out rows [0, (S-CP)*B)
    const int Mdec = (kS - kCP) * kB;                  // 1024
    launch_gemm(src + (size_t)kCP * kB * kD, dec_w1, dec_b1, ff,
                Mdec, kNHID, kD, kD, kD, kNHID, 1, stream);
    launch_gemm(ff, dec_w2, dec_b2, out,
                Mdec, kNOUT, kNHID, kNHID, kNHID, kNOUT, 0, stream);

    // dp head: src rows [S, T) -> out rows [(S-CP)*B, ...)
    const int Mdp = kP * kB;                           // 512
    launch_gemm(src + (size_t)kS * kB * kD, dp_w1, dp_b1, ff,
                Mdp, kNHID, kD, kD, kD, kNHID, 2, stream);
    launch_gemm(ff, dp_w2, dp_b2, out + (size_t)Mdec * kNOUT,
                Mdp, kNOUT, kNHID, kNHID, kNHID, kNOUT, 0, stream);
}